// Topology_41755672051931
// MI455X (gfx1250) — compile-verified
//
#include <hip/hip_runtime.h>
#include <stdint.h>

// ---------------------------------------------------------------------------
// GLI "topology" distance between two motions.
// VALU/arcsin-bound, no GEMM structure (K=3) -> WMMA inapplicable.
// CDNA5 paths used: wave32 frame-per-wave mapping, async global->LDS staging
// (ASYNCcnt), LDS float atomics, split-counter wait.
// ---------------------------------------------------------------------------

#define FRAMES 16384
#define JOINTS 22
#define FPF    (JOINTS * 3)   // 66 floats per frame per motion
#define FPB    8              // frames per block (8 waves of 32 = 256 threads)
#define NSEG   16
#define NBIN   25
#define INV4PI 0.07957747154594767f

// Segment tables: PATHS = [[2,5,8,11],[1,4,7,10],[3,6,9,12,15],[14,17,19,21],[13,16,18,20]]
__constant__ int c_ss[NSEG]  = {2,5,8,  1,4,7,  3,6,9,12,  14,17,19, 13,16,18};
__constant__ int c_se[NSEG]  = {5,8,11, 4,7,10, 6,9,12,15, 17,19,21, 16,18,20};
__constant__ int c_sp[NSEG]  = {0,0,0,  1,1,1,  2,2,2,2,   3,3,3,    4,4,4};
__constant__ int c_sp5[NSEG] = {0,0,0,  5,5,5,  10,10,10,10, 15,15,15, 20,20,20};

struct V3 { float x, y, z; };

__device__ __forceinline__ V3 v3sub(V3 a, V3 b) { return V3{a.x-b.x, a.y-b.y, a.z-b.z}; }
__device__ __forceinline__ V3 v3cross(V3 a, V3 b) {
    return V3{a.y*b.z - a.z*b.y, a.z*b.x - a.x*b.z, a.x*b.y - a.y*b.x};
}
__device__ __forceinline__ float v3dot(V3 a, V3 b) { return a.x*b.x + a.y*b.y + a.z*b.z; }
__device__ __forceinline__ V3 safenorm(V3 v) {
    float n2 = v3dot(v, v);
    if (n2 > 0.0f) {
        float inv = 1.0f / sqrtf(n2);
        return V3{v.x*inv, v.y*inv, v.z*inv};
    }
    return V3{0.0f, 0.0f, 0.0f};
}
__device__ __forceinline__ float clamp1(float d) { return fminf(1.0f, fmaxf(-1.0f, d)); }
__device__ __forceinline__ V3 ld3(const float* f, int j) { return V3{f[3*j], f[3*j+1], f[3*j+2]}; }

// Address-space helper types for gfx1250 async-to-LDS builtin (typed int*).
typedef __attribute__((address_space(1))) int as1i;
typedef __attribute__((address_space(3))) int as3i;

__global__ __launch_bounds__(256) void gli_frames_kernel(
    const float* __restrict__ m1, const float* __restrict__ m2,
    float* __restrict__ gliw, int* __restrict__ flags)
{
    __shared__ float s1b[FPB * FPF];
    __shared__ float s2b[FPB * FPF];
    __shared__ float bins[FPB * NBIN];

    const int tid = threadIdx.x;
    const size_t base = (size_t)blockIdx.x * (FPB * FPF);

    // ---- Stage 8 frames of both motions into LDS -------------------------
#if defined(__gfx1250__) && __has_builtin(__builtin_amdgcn_global_load_async_to_lds_b32)
    for (int t = tid; t < FPB * FPF; t += 256) {
        __builtin_amdgcn_global_load_async_to_lds_b32(
            (as1i*)(uintptr_t)(m1 + base + t),
            (as3i*)(uint32_t)(uintptr_t)&s1b[t], 0, 0);
        __builtin_amdgcn_global_load_async_to_lds_b32(
            (as1i*)(uintptr_t)(m2 + base + t),
            (as3i*)(uint32_t)(uintptr_t)&s2b[t], 0, 0);
    }
#if __has_builtin(__builtin_amdgcn_s_wait_asynccnt)
    __builtin_amdgcn_s_wait_asynccnt(0);
#else
    asm volatile("s_wait_asynccnt 0" ::: "memory");
#endif
#else
    for (int t = tid; t < FPB * FPF; t += 256) {
        s1b[t] = m1[base + t];
        s2b[t] = m2[base + t];
    }
#endif
    if (tid < FPB * NBIN) bins[tid] = 0.0f;
    __syncthreads();

    // ---- One wave32 per frame; lane owns motion2 segment j = lane&15 -----
    const int w    = tid >> 5;         // wave (frame slot) 0..7
    const int lane = tid & 31;
    const float* F1 = &s1b[w * FPF];
    const float* F2 = &s2b[w * FPF];

    const int j  = lane & 15;
    const int pb = c_sp[j];
    const V3 S2  = ld3(F2, c_ss[j]);
    const V3 E2  = ld3(F2, c_se[j]);
    const V3 R34 = v3sub(E2, S2);

    const int ibase = lane >> 4;       // 0 or 1; i = ibase + 2k covers all 16 i
#pragma unroll
    for (int k = 0; k < 8; ++k) {
        const int i = ibase + 2 * k;
        const V3 s1 = ld3(F1, c_ss[i]);
        const V3 e1 = ld3(F1, c_se[i]);

        const V3 r12 = v3sub(e1, s1);
        const V3 r13 = v3sub(S2, s1);
        const V3 r14 = v3sub(E2, s1);
        const V3 r23 = v3sub(S2, e1);
        const V3 r24 = v3sub(E2, e1);

        const V3 n0 = safenorm(v3cross(r13, r14));
        const V3 n1 = safenorm(v3cross(r14, r24));
        const V3 n2 = safenorm(v3cross(r24, r23));
        const V3 n3 = safenorm(v3cross(r23, r13));

        float g = asinf(clamp1(v3dot(n0, n1)))
                + asinf(clamp1(v3dot(n1, n2)))
                + asinf(clamp1(v3dot(n2, n3)))
                + asinf(clamp1(v3dot(n3, n0)));

        const float sg = v3dot(v3cross(R34, r12), r13);
        g *= (sg > 0.0f) ? INV4PI : -INV4PI;

        atomicAdd(&bins[w * NBIN + c_sp5[i] + pb], g);   // ds_add_f32
    }

    // ---- Bbox-overlap window flag for this frame (lane 0, data in LDS) ---
    if (lane == 0) {
        float x1n = F1[0], x1x = F1[0], z1n = F1[2], z1x = F1[2];
        float x2n = F2[0], x2x = F2[0], z2n = F2[2], z2x = F2[2];
        for (int q = 1; q < JOINTS; ++q) {
            float a = F1[3*q], b = F1[3*q + 2];
            x1n = fminf(x1n, a); x1x = fmaxf(x1x, a);
            z1n = fminf(z1n, b); z1x = fmaxf(z1x, b);
            float c = F2[3*q], d = F2[3*q + 2];
            x2n = fminf(x2n, c); x2x = fmaxf(x2x, c);
            z2n = fminf(z2n, d); z2x = fmaxf(z2x, d);
        }
        const bool ovx = !((x1x < x2n) || (x2x < x1n));
        const bool ovz = !((z1x < z2n) || (z2x < z1n));
        flags[blockIdx.x * FPB + w] = (ovx && ovz) ? 1 : 0;
    }
    __syncthreads();

    // ---- Spill the 8x25 per-frame GLI matrices to workspace --------------
    for (int t = tid; t < FPB * NBIN; t += 256) {
        gliw[(size_t)blockIdx.x * (FPB * NBIN) + t] = bins[t];
    }
}

__global__ __launch_bounds__(256) void vel_kernel(
    const float* __restrict__ gliw, const int* __restrict__ flags,
    float* __restrict__ out)
{
    const int f = blockIdx.x * blockDim.x + threadIdx.x;
    if (f >= FRAMES - 1) return;

    const int fm1 = (f > 0) ? flags[f - 1] : 0;
    const int f0  = flags[f];
    const int f1  = flags[f + 1];
    const int f2  = (f + 2 < FRAMES) ? flags[f + 2] : 0;
    const float ma = (fm1 | f0 | f1) ? 1.0f : 0.0f;   // mask for frame f
    const float mb = (f0 | f1 | f2) ? 1.0f : 0.0f;    // mask for frame f+1

    const float* a = gliw + (size_t)f * NBIN;
    const float* b = a + NBIN;
    float v = 0.0f;
#pragma unroll
    for (int q = 0; q < NBIN; ++q) {
        v = fmaxf(v, fabsf(b[q] * mb - a[q] * ma));
    }
    out[f] = v;
}

extern "C" void kernel_launch(void* const* d_in, const int* in_sizes, int n_in,
                              void* d_out, int out_size, void* d_ws, size_t ws_size,
                              hipStream_t stream) {
    const float* m1 = (const float*)d_in[0];
    const float* m2 = (const float*)d_in[1];
    float* out = (float*)d_out;

    float* gliw = (float*)d_ws;                                        // FRAMES*25 floats
    int*   flgs = (int*)((char*)d_ws + (size_t)FRAMES * NBIN * sizeof(float)); // FRAMES ints

    gli_frames_kernel<<<FRAMES / FPB, 256, 0, stream>>>(m1, m2, gliw, flgs);
    vel_kernel<<<(FRAMES - 1 + 255) / 256, 256, 0, stream>>>(gliw, flgs, out);
}